// ActorNetSpiking_38113539784776
// MI455X (gfx1250) — compile-verified
//
#include <hip/hip_runtime.h>
#include <math.h>

// ---------------------------------------------------------------------------
// ActorNetSpiking forward on MI455X (gfx1250), fp32 WMMA 16x16x4 path.
// 4 LIF layers x 50 timesteps; one fused GEMM+LIF+reduce kernel per layer-step.
// Each wave owns TWO M-adjacent 16x16 C tiles (two independent WMMA chains
// sharing one B fragment); 4 waves/block share an LDS-staged B panel.
// ---------------------------------------------------------------------------

typedef float v2f __attribute__((ext_vector_type(2)));
typedef float v8f __attribute__((ext_vector_type(8)));

#define BSZ    2048
#define SDIM   256
#define HDIM   256
#define ADIM   2
#define TSTEPS 50
#define KDIM   256
#define BH     (BSZ * HDIM)       // 524288
#define BA     (BSZ * ADIM)       // 4096
#define SUMSCALE 16777216.0f      // 2^24 fixed-point quantum for deterministic sums

// Monotone order-preserving float <-> uint key (for atomicMax/Min).
__device__ __forceinline__ unsigned f2key(float f) {
  unsigned u = __float_as_uint(f);
  return (u & 0x80000000u) ? ~u : (u | 0x80000000u);
}
__device__ __forceinline__ float key2f(unsigned k) {
  return (k & 0x80000000u) ? __uint_as_float(k & 0x7fffffffu)
                           : __uint_as_float(~k);
}

// Slot layout (8 x 32-bit words, 8B aligned):
// [0:1] i64 sum(v)*2^24   [2:3] i64 sum(vth)*2^24
// [4] maxkey(v) [5] minkey(v) [6] maxkey(vth) [7] minkey(vth)

__global__ void copy_f32(float* __restrict__ dst, const float* __restrict__ src, int n) {
  int i = blockIdx.x * blockDim.x + threadIdx.x;
  if (i < n) dst[i] = src[i];
}

__global__ void init_misc(float* __restrict__ acc, float* __restrict__ slots, int nSlots) {
  int i = blockIdx.x * blockDim.x + threadIdx.x;
  if (i < BA) acc[i] = 0.0f;
  if (i < nSlots) {
    unsigned long long* cs = (unsigned long long*)(slots + (size_t)i * 8);
    cs[0] = 0ull; cs[1] = 0ull;
    unsigned* cu = (unsigned*)(slots + (size_t)i * 8);
    cu[4] = f2key(-INFINITY); cu[5] = f2key(INFINITY);
    cu[6] = f2key(-INFINITY); cu[7] = f2key(INFINITY);
  }
}

__global__ void finalize_out(float* __restrict__ out, const float* __restrict__ acc, int n) {
  int i = blockIdx.x * blockDim.x + threadIdx.x;
  if (i < n) out[i] = acc[i] * (1.0f / (float)TSTEPS);
}

// One fused layer-step:  u = u*0.5 + A@W.T + b ; v = v_prev*0.75*(1-s_prev) + u ;
// temporal recomputed inline from v_prev + prev-step global scalars; vth/spike;
// wave-level reduce of v/vth (sum/max/min) -> per-(step,layer) slot via atomics.
//
// Template params: AROWS/ACOLS = A row/col element strides (compile-time so the
// spike layers get b64 A loads), NTN = N tiles, NV = valid output columns.
template <int AROWS, int ACOLS, int NTN, int NV>
__global__ void __launch_bounds__(128)
lif_gemm_step(const float* __restrict__ A, int aOffset,
              const float* __restrict__ W, const float* __restrict__ bias,
              float* __restrict__ u, float* __restrict__ v, float* __restrict__ s,
              float* __restrict__ accSpike,
              const float* __restrict__ prevSlot, float* __restrict__ curSlot,
              int isFirst)
{
  constexpr float invCount = 1.0f / (float)(BSZ * NV);
  constexpr int   LDSW     = KDIM + 4;          // +4 floats: distinct banks per column
  __shared__ float ldsW[16 * LDSW];

  const int waveId = threadIdx.x >> 5;
  const int lane   = threadIdx.x & 31;
  const int tileN  = (NTN == 1) ? 0 : (int)(blockIdx.x % NTN);
  const int mGroup = (NTN == 1) ? (int)blockIdx.x : (int)(blockIdx.x / NTN);
  const int tileM0 = (mGroup * 4 + waveId) * 2; // two M-adjacent tiles per wave
  const int row0   = tileM0 << 4;               // rows row0..row0+31
  const int col0   = tileN << 4;
  const int half   = lane >> 4;                 // lanes 16-31: K+2..K+3
  const int mn     = lane & 15;                 // M (A) / N (B,C) index

  // Cooperative stage of the 16x256 B panel (W rows col0..col0+15, row-clamped
  // for the N=2 layer) into LDS, once per block.
  for (int i = threadIdx.x; i < 16 * KDIM; i += 128) {
    const int c    = i >> 8;                    // KDIM == 256
    const int k    = i & (KDIM - 1);
    const int gcol = col0 + c;
    const int sc   = (NV >= 16) ? gcol : ((gcol < NV) ? gcol : (NV - 1));
    ldsW[c * LDSW + k] = W[(size_t)sc * KDIM + k];
  }
  __syncthreads();

  const int  bcol   = col0 + mn;
  const bool cValid = (NV >= 16) || (bcol < NV);
  const float* __restrict__ abase0 = A + (size_t)(row0 + mn) * AROWS + aOffset;
  const float* __restrict__ abase1 = abase0 + (size_t)16 * AROWS;
  const float* __restrict__ lbase  = ldsW + mn * LDSW;
  __builtin_prefetch(abase0, 0, 3);
  __builtin_prefetch(abase1, 0, 3);

  v8f acc0 = {0.f, 0.f, 0.f, 0.f, 0.f, 0.f, 0.f, 0.f};
  v8f acc1 = {0.f, 0.f, 0.f, 0.f, 0.f, 0.f, 0.f, 0.f};
#pragma unroll 8
  for (int k = 0; k < KDIM; k += 4) {
    const int kk = k + 2 * half;
    v2f a0, a1, b;
    if (ACOLS == 1) {
      a0 = *(const v2f*)(abase0 + kk);          // contiguous: single b64 load
      a1 = *(const v2f*)(abase1 + kk);
    } else {
      a0.x = abase0[(size_t)kk * ACOLS];        // x is [B,S,T]: stride-T columns
      a0.y = abase0[(size_t)(kk + 1) * ACOLS];
      a1.x = abase1[(size_t)kk * ACOLS];
      a1.y = abase1[(size_t)(kk + 1) * ACOLS];
    }
    b = *(const v2f*)(lbase + kk);              // shared B fragment from LDS
    // Two independent accumulation chains -> WMMA latency overlap.
    acc0 = __builtin_amdgcn_wmma_f32_16x16x4_f32(false, a0, false, b,
                                                 (short)0, acc0, false, false);
    acc1 = __builtin_amdgcn_wmma_f32_16x16x4_f32(false, a1, false, b,
                                                 (short)0, acc1, false, false);
  }

  // Previous-step global scalars (uniform per kernel).
  float Vm_p = 0.0f, Vt_p = 0.0f;
  if (!isFirst) {
    const long long* ps = (const long long*)prevSlot;
    const unsigned*  pu = (const unsigned*)prevSlot;
    const float mean_v = (float)ps[0] * (1.0f / SUMSCALE) * invCount;
    const float mean_t = (float)ps[1] * (1.0f / SUMSCALE) * invCount;
    Vm_p = mean_v - 0.2f * (key2f(pu[4]) - key2f(pu[5]));
    Vt_p = mean_t - 0.2f * (key2f(pu[6]) - key2f(pu[7]));
  }

  float sum_v = 0.0f, sum_t = 0.0f;
  float max_v = -INFINITY, min_v = INFINITY;
  float max_t = -INFINITY, min_t = INFINITY;
  const float bias_c = cValid ? bias[bcol] : 0.0f;

#pragma unroll
  for (int pp = 0; pp < 2; ++pp) {
    const int rbase = row0 + 16 * pp + 8 * half;
#pragma unroll
    for (int r = 0; r < 8; ++r) {
      const float gemm = (pp == 0) ? acc0[r] : acc1[r];
      const int row = rbase + r;                // C layout: VGPR r -> M=r / r+8
      if (cValid) {
        const size_t idx = (size_t)row * NV + bcol;
        const float uo = u[idx];
        const float vo = v[idx];
        const float so = s[idx];
        const float un = uo * 0.5f + gemm + bias_c;            // u*CDECAY + A@W.T + b
        const float vn = vo * 0.75f * (1.0f - so) + un;        // v*VDECAY*(1-s) + u
        float temporal;
        if (isFirst) {
          temporal = 0.5f;                                     // VTH_INIT
        } else {                                               // deferred temporal update
          const float z = vo - Vm_p;
          temporal = 0.01f * z + Vt_p + logf(1.0f + expf(z * 0.25f));
        }
        const float energy = expf((vo - vn) * (1.0f / 3.0f)) - 1.0f;
        const float vth    = 0.5f * temporal + 0.5f * energy;
        const float sn     = (vn > vth) ? 1.0f : 0.0f;
        u[idx] = un; v[idx] = vn; s[idx] = sn;
        if (NV == ADIM) accSpike[idx] += sn;                   // layer-4 spike acc
        sum_v += vn;                sum_t += vth;
        max_v  = fmaxf(max_v, vn);  min_v  = fminf(min_v, vn);
        max_t  = fmaxf(max_t, vth); min_t  = fminf(min_t, vth);
      }
    }
  }

  // Wave32 butterfly reduction.
#pragma unroll
  for (int off = 16; off > 0; off >>= 1) {
    sum_v += __shfl_xor(sum_v, off, 32);
    sum_t += __shfl_xor(sum_t, off, 32);
    max_v = fmaxf(max_v, __shfl_xor(max_v, off, 32));
    min_v = fminf(min_v, __shfl_xor(min_v, off, 32));
    max_t = fmaxf(max_t, __shfl_xor(max_t, off, 32));
    min_t = fminf(min_t, __shfl_xor(min_t, off, 32));
  }

  if (lane == 0) {
    unsigned long long* cs = (unsigned long long*)curSlot;
    atomicAdd(&cs[0], (unsigned long long)(long long)llrintf(sum_v * SUMSCALE));
    atomicAdd(&cs[1], (unsigned long long)(long long)llrintf(sum_t * SUMSCALE));
    unsigned* cu = (unsigned*)curSlot;
    atomicMax(&cu[4], f2key(max_v));
    atomicMin(&cu[5], f2key(min_v));
    atomicMax(&cu[6], f2key(max_t));
    atomicMin(&cu[7], f2key(min_t));
  }
}

extern "C" void kernel_launch(void* const* d_in, const int* in_sizes, int n_in,
                              void* d_out, int out_size, void* d_ws, size_t ws_size,
                              hipStream_t stream) {
  (void)in_sizes; (void)n_in; (void)out_size; (void)ws_size;

  const float* x  = (const float*)d_in[0];
  const float* Wm[4] = {(const float*)d_in[13], (const float*)d_in[15],
                        (const float*)d_in[17], (const float*)d_in[19]};
  const float* bm[4] = {(const float*)d_in[14], (const float*)d_in[16],
                        (const float*)d_in[18], (const float*)d_in[20]};

  // Workspace layout (floats).
  float* p = (float*)d_ws;
  float *U[4], *V[4], *S[4];
  for (int l = 0; l < 3; ++l) {
    U[l] = p + (size_t)(3 * l + 0) * BH;
    V[l] = p + (size_t)(3 * l + 1) * BH;
    S[l] = p + (size_t)(3 * l + 2) * BH;
  }
  U[3] = p + (size_t)9 * BH;
  V[3] = U[3] + BA;
  S[3] = U[3] + 2 * BA;
  float* accS  = U[3] + 3 * BA;
  float* slots = accS + BA;                    // TSTEPS*4 slots x 8 words

  // Initialize states from inputs (do not mutate d_in), zero acc, init slots.
  for (int l = 0; l < 4; ++l) {
    const int n = (l == 3) ? BA : BH;
    const int blocks = (n + 255) / 256;
    copy_f32<<<blocks, 256, 0, stream>>>(U[l], (const float*)d_in[1 + 3 * l + 0], n);
    copy_f32<<<blocks, 256, 0, stream>>>(V[l], (const float*)d_in[1 + 3 * l + 1], n);
    copy_f32<<<blocks, 256, 0, stream>>>(S[l], (const float*)d_in[1 + 3 * l + 2], n);
  }
  init_misc<<<(BA + 255) / 256, 256, 0, stream>>>(accS, slots, TSTEPS * 4);

  // 50 timesteps x 4 layers, fully fused; each kernel reads prev-step slot scalars.
  const int blocksH = (BSZ / 128) * (HDIM / 16);  // 16 M-groups x 16 N-tiles = 256
  const int blocksA = (BSZ / 128);                // 16
  for (int t = 0; t < TSTEPS; ++t) {
    const int first = (t == 0) ? 1 : 0;
    float* sl;
    const float* pv;
#define SLOTS(tt, ll) (slots + (size_t)((tt) * 4 + (ll)) * 8)
    sl = SLOTS(t, 0); pv = first ? sl : SLOTS(t - 1, 0);
    lif_gemm_step<SDIM * TSTEPS, TSTEPS, HDIM / 16, HDIM>
        <<<blocksH, 128, 0, stream>>>(x, t, Wm[0], bm[0],
                                      U[0], V[0], S[0], accS, pv, sl, first);
    sl = SLOTS(t, 1); pv = first ? sl : SLOTS(t - 1, 1);
    lif_gemm_step<HDIM, 1, HDIM / 16, HDIM>
        <<<blocksH, 128, 0, stream>>>(S[0], 0, Wm[1], bm[1],
                                      U[1], V[1], S[1], accS, pv, sl, first);
    sl = SLOTS(t, 2); pv = first ? sl : SLOTS(t - 1, 2);
    lif_gemm_step<HDIM, 1, HDIM / 16, HDIM>
        <<<blocksH, 128, 0, stream>>>(S[1], 0, Wm[2], bm[2],
                                      U[2], V[2], S[2], accS, pv, sl, first);
    sl = SLOTS(t, 3); pv = first ? sl : SLOTS(t - 1, 3);
    lif_gemm_step<HDIM, 1, 1, ADIM>
        <<<blocksA, 128, 0, stream>>>(S[2], 0, Wm[3], bm[3],
                                      U[3], V[3], S[3], accS, pv, sl, first);
#undef SLOTS
  }

  finalize_out<<<(BA + 255) / 256, 256, 0, stream>>>((float*)d_out, accS, BA);
}